// EdgeConv_66391604462124
// MI455X (gfx1250) — compile-verified
//
#include <hip/hip_runtime.h>

// ---------------------------------------------------------------------------
// EdgeConv (DGCNN) pipeline for MI455X (gfx1250), wave32, f32 WMMA 16x16x4.
//   A) transpose x (B,C,N)->xT (B,N,C) + per-point squared norms
//   B) kNN: tiled d2 = |xn|^2+|xm|^2-2*X.X^T via v_wmma_f32_16x16x4_f32,
//      double-buffered async L2->LDS chunk streaming (ASYNCcnt) via
//      global_load_async_to_lds_b128; top-20 per row.
//   C) edge GEMM pass1: y = W.edge + b, per-channel sum/sumsq
//   D) finalize BN scale/shift
//   E) edge GEMM pass2: recompute y, BN+ReLU, k-max via uint atomicMax
// ---------------------------------------------------------------------------

typedef float v2f __attribute__((ext_vector_type(2)));
typedef float v8f __attribute__((ext_vector_type(8)));

static constexpr int   Bs   = 8;
static constexpr int   Cs   = 64;
static constexpr int   Ns   = 4096;
static constexpr int   Ks   = 20;
static constexpr int   Os   = 64;
static constexpr int   C2   = 2 * Cs;    // 128
static constexpr float EPSF = 1e-5f;

#if defined(__has_builtin)
#if __has_builtin(__builtin_amdgcn_global_load_async_to_lds_b128) && \
    __has_builtin(__builtin_amdgcn_s_wait_asynccnt)
#define USE_ASYNC_LDS 1
#endif
#endif
#ifndef USE_ASYNC_LDS
#define USE_ASYNC_LDS 0
#endif

#if USE_ASYNC_LDS
// Parameter type per hipcc diagnostic: pointer to 16-byte int vector, global
// (AS1) source / LDS (AS3) destination.
typedef int b128_t __attribute__((vector_size(4 * sizeof(int))));
typedef __attribute__((address_space(1))) b128_t as1_b128;
typedef __attribute__((address_space(3))) b128_t as3_b128;
#endif

static __device__ __forceinline__ v8f wmma_f32_16x16x4(v2f a, v2f b, v8f c) {
  // D = A(16x4) * B(4x16) + C(16x16), fp32, wave32.
  return __builtin_amdgcn_wmma_f32_16x16x4_f32(false, a, false, b, (short)0, c,
                                               false, false);
}

// ---------------------------------------------------------------------------
// Kernel A: LDS-tiled transpose x(B,C,N) -> xT(B,N,C), plus sq[b][n] = |x_n|^2
// grid (N/64, B) x 256 threads
// ---------------------------------------------------------------------------
__global__ __launch_bounds__(256) void xt_sq_kernel(const float* __restrict__ x,
                                                    float* __restrict__ xT,
                                                    float* __restrict__ sq) {
  __shared__ float lX[64][65];   // +1 pad: conflict-free column reads
  __shared__ float ps[64][4];
  const int b  = blockIdx.y;
  const int n0 = blockIdx.x * 64;
  const int t  = threadIdx.x;
  const int nl = t & 63;
  const int cg = t >> 6;                 // 0..3
  for (int i = 0; i < 16; ++i) {
    const int c = cg * 16 + i;
    lX[c][nl] = x[(size_t)b * Cs * Ns + (size_t)c * Ns + (n0 + nl)];
  }
  __syncthreads();
  const int row = t >> 2;                // 0..63 (local n)
  const int cq  = t & 3;                 // 16-wide c chunk
  float s = 0.f;
  for (int j = 0; j < 16; ++j) {
    const float v = lX[cq * 16 + j][row];
    xT[(size_t)b * Ns * Cs + (size_t)(n0 + row) * Cs + (cq * 16 + j)] = v;
    s += v * v;
  }
  ps[row][cq] = s;
  __syncthreads();
  if (cq == 0)
    sq[b * Ns + n0 + row] = ps[row][0] + ps[row][1] + ps[row][2] + ps[row][3];
}

// ---------------------------------------------------------------------------
// Kernel B: kNN. One WG = 32 query rows of one batch. Streams N in 64-wide
// column chunks, double-buffered: chunk i+1 is copied L2->LDS with
// global_load_async_to_lds_b128 while chunk i is consumed by WMMA + top-k.
// 8 waves each own one 16x16 tile of X.X^T per chunk (K=64 -> 16 wmma).
// grid (N/32, B) x 256 threads.  LDS ~48.4 KB (>=2 WGs resident per WGP).
// ---------------------------------------------------------------------------
static constexpr int CW = 64;           // chunk width (candidate points)

__global__ __launch_bounds__(256) void knn_kernel(const float* __restrict__ xT,
                                                  const float* __restrict__ sq,
                                                  int* __restrict__ idxout) {
  __shared__ float lA[32][64];          // query point features
  __shared__ float lB[2][CW][64];       // double-buffered candidate chunks
  __shared__ float lD[32][CW];          // d2 tile
  __shared__ float lsqn[32];
  __shared__ float lsqm[CW];

  const int b    = blockIdx.y;
  const int r0   = blockIdx.x * 32;
  const int t    = threadIdx.x;
  const int lane = t & 31;
  const int wave = t >> 5;              // 0..7
  const int half = lane >> 4;           // K-half selector for f32 fragments
  const int l15  = lane & 15;
  const int mt   = wave >> 2;           // row-tile 0..1
  const int colb = (wave & 3) * 16;     // col-tile base 0..48

  const float* srcA = xT + (size_t)b * Ns * Cs;   // this batch's point rows

  for (int i = t; i < 32 * 64; i += 256)
    lA[i >> 6][i & 63] = srcA[(size_t)r0 * Cs + i];
  if (t < 32) lsqn[t] = sq[b * Ns + r0 + t];

  float bd[Ks];
  int   bi[Ks];
  for (int i = 0; i < Ks; ++i) { bd[i] = 3.4e38f; bi[i] = 0; }

  constexpr int NCH = Ns / CW;          // 64 chunks

#if USE_ASYNC_LDS
  // One chunk = CW*64 floats = 16 KB = 1024 b128 slots; 4 async b128 per
  // thread = 4 async instructions per wave.
  {
    const char* g = (const char*)srcA;  // chunk 0 at offset 0
    char*       l = (char*)&lB[0][0][0];
    for (int i = 0; i < 4; ++i) {
      const int idx = (t + i * 256) * 16;
      __builtin_amdgcn_global_load_async_to_lds_b128(
          (as1_b128*)(g + idx), (as3_b128*)(l + idx), 0, 0);
    }
  }
#endif

  for (int ch = 0; ch < NCH; ++ch) {
    const int cur = ch & 1;
    const int m0  = ch * CW;

#if USE_ASYNC_LDS
    if (ch + 1 < NCH) {                 // prefetch next chunk into other buf
      const char* g = (const char*)(srcA + (size_t)(m0 + CW) * Cs);
      char*       l = (char*)&lB[cur ^ 1][0][0];
      for (int i = 0; i < 4; ++i) {
        const int idx = (t + i * 256) * 16;
        __builtin_amdgcn_global_load_async_to_lds_b128(
            (as1_b128*)(g + idx), (as3_b128*)(l + idx), 0, 0);
      }
    }
    if (t < CW) lsqm[t] = sq[b * Ns + m0 + t];
    if (ch + 1 < NCH) __builtin_amdgcn_s_wait_asynccnt(4);  // cur buf done
    else              __builtin_amdgcn_s_wait_asynccnt(0);
    __syncthreads();
#else
    __syncthreads();                    // lB/lD safe to overwrite
    for (int i = t; i < CW * 64; i += 256)
      lB[cur][i >> 6][i & 63] = srcA[(size_t)m0 * Cs + i];
    if (t < CW) lsqm[t] = sq[b * Ns + m0 + t];
    __syncthreads();
#endif

    // each wave: one 16x16 tile of X.X^T, K=64 in 16 wmma steps
    v8f acc = {};
    for (int kk = 0; kk < 16; ++kk) {
      const int kb = kk * 4 + half * 2;
      v2f a, bb;
      a.x  = lA[mt * 16 + l15][kb];
      a.y  = lA[mt * 16 + l15][kb + 1];
      bb.x = lB[cur][colb + l15][kb];   // B[k][m] = x_m[k]
      bb.y = lB[cur][colb + l15][kb + 1];
      acc  = wmma_f32_16x16x4(a, bb, acc);
    }
    {
      const int   col = colb + l15;
      const float sm  = lsqm[col];
      for (int r = 0; r < 8; ++r) {
        const int M = mt * 16 + r + half * 8;   // C/D layout (ISA 7.12.2)
        lD[M][col]  = lsqn[M] + sm - 2.0f * acc[r];
      }
    }
    __syncthreads();

    if (t < 32) {                       // owner lane scans its row
      const int n = r0 + t;
      for (int j = 0; j < CW; ++j) {
        const int m = m0 + j;
        if (m == n) continue;           // exclude self
        const float d = lD[t][j];
        if (d < bd[Ks - 1]) {
          int p = Ks - 1;
          while (p > 0 && bd[p - 1] > d) {
            bd[p] = bd[p - 1]; bi[p] = bi[p - 1]; --p;
          }
          bd[p] = d; bi[p] = m;
        }
      }
    }
    __syncthreads();                    // scan done before lD/lB overwrite
  }
  if (t < 32) {
    const int n = r0 + t;
    for (int i = 0; i < Ks; ++i) idxout[(b * Ns + n) * Ks + i] = bi[i];
  }
}

// ---------------------------------------------------------------------------
// Kernels C/E: edge GEMM  y[r][o] = W[o,:].edge[r] + bias[o]
//   edge rows r = (b,n,k) flattened, M_total = B*N*K = 655360, K-dim = 128.
// One WG = 64 rows x 64 outputs = 16 WMMA tiles; 8 waves x 2 tiles each.
// W staged in two 64-K halves (16 KB) so static LDS stays ~48.5 KB.
// PASS==1: per-channel sum/sumsq (LDS -> global atomics).
// PASS==2: yn = scale*y+shift, ReLU, k-max folded via uint atomicMax.
// grid 10240 x 256 threads.
// ---------------------------------------------------------------------------
template <int PASS>
__global__ __launch_bounds__(256) void edge_kernel(
    const float* __restrict__ xT, const int* __restrict__ nbr,
    const float* __restrict__ W, const float* __restrict__ bias,
    float* __restrict__ stats,          // PASS1: [sum[64], sumsq[64]]
    const float* __restrict__ ss,       // PASS2: [scale[64], shift[64]]
    unsigned* __restrict__ out) {       // PASS2: (B,O,N) as uint bits
  __shared__ float lW[64][64];          // one K-half of W, [k][o]
  __shared__ float lE[64][128];         // edge rows
  __shared__ float lsum[64], lsq[64];

  const int    t    = threadIdx.x;
  const int    lane = t & 31;
  const int    wave = t >> 5;
  const int    half = lane >> 4;
  const int    l15  = lane & 15;
  const size_t R0   = (size_t)blockIdx.x * 64;

  {  // build 64 edge rows: 4 threads/row, 32 features each
    const int    lr  = t >> 2;
    const int    prt = t & 3;
    const size_t r   = R0 + lr;
    const int    b   = (int)(r / ((size_t)Ns * Ks));
    const int    rem = (int)(r % ((size_t)Ns * Ks));
    const int    n   = rem / Ks;
    const int    k   = rem % Ks;
    const int    nb  = nbr[(b * Ns + n) * Ks + k];
    const float* xc  = xT + ((size_t)b * Ns + n) * Cs;
    const float* xn  = xT + ((size_t)b * Ns + nb) * Cs;
    for (int j = 0; j < 32; ++j) {
      const int c = prt * 32 + j;
      lE[lr][c]   = (c < Cs) ? xc[c] : (xn[c - Cs] - xc[c - Cs]);
    }
  }
  if (t < 64) { lsum[t] = 0.f; lsq[t] = 0.f; }

  // 16 tiles (mt 0..3, ct 0..3); wave owns tile ids {wave, wave+8}
  const int mt0 = wave & 3, ct0 = wave >> 2;
  const int mt1 = (wave + 8) & 3, ct1 = (wave + 8) >> 2;
  v8f acc0 = {}, acc1 = {};

  for (int kh = 0; kh < 2; ++kh) {
    __syncthreads();                       // lE ready (kh=0) / lW reusable
    for (int i = t; i < 64 * 64; i += 256) // stage W K-half, lW[k][o]=W[o][k]
      lW[i >> 6][i & 63] = W[(i & 63) * C2 + kh * 64 + (i >> 6)];
    __syncthreads();
    for (int kk = 0; kk < 16; ++kk) {
      const int kb = kk * 4 + half * 2;
      const int ke = kh * 64 + kb;
      v2f a0, a1, b0, b1;
      a0.x = lE[mt0 * 16 + l15][ke];  a0.y = lE[mt0 * 16 + l15][ke + 1];
      b0.x = lW[kb][ct0 * 16 + l15];  b0.y = lW[kb + 1][ct0 * 16 + l15];
      acc0 = wmma_f32_16x16x4(a0, b0, acc0);
      a1.x = lE[mt1 * 16 + l15][ke];  a1.y = lE[mt1 * 16 + l15][ke + 1];
      b1.x = lW[kb][ct1 * 16 + l15];  b1.y = lW[kb + 1][ct1 * 16 + l15];
      acc1 = wmma_f32_16x16x4(a1, b1, acc1);
    }
  }
  __syncthreads();

  // tile epilogues
  for (int which = 0; which < 2; ++which) {
    const v8f acc = which ? acc1 : acc0;
    const int mt  = which ? mt1 : mt0;
    const int ct  = which ? ct1 : ct0;
    const int o   = ct * 16 + l15;         // lane's output channel
    const float bv = bias[o];
    if (PASS == 1) {
      float s = 0.f, q = 0.f;
      for (int r = 0; r < 8; ++r) {
        const float y = acc[r] + bv;
        s += y; q += y * y;
      }
      atomicAdd(&lsum[o], s);
      atomicAdd(&lsq[o], q);
    } else {
      const float sc = ss[o], sh = ss[64 + o];
      for (int r = 0; r < 8; ++r) {
        const float  y  = acc[r] + bv;
        const float  v  = fmaxf(sc * y + sh, 0.f);     // BN + ReLU, v >= 0
        const size_t rg = R0 + (size_t)(mt * 16 + r + half * 8);
        const int    b  = (int)(rg / ((size_t)Ns * Ks));
        const int    n  = (int)(rg % ((size_t)Ns * Ks)) / Ks;
        atomicMax(&out[((size_t)b * Os + o) * Ns + n], __float_as_uint(v));
      }
    }
  }
  if (PASS == 1) {
    __syncthreads();
    if (t < 64) {
      atomicAdd(&stats[t], lsum[t]);
      atomicAdd(&stats[64 + t], lsq[t]);
    }
  }
}

// ---------------------------------------------------------------------------
// Kernel D: BN finalize -> scale/shift
// ---------------------------------------------------------------------------
__global__ void finalize_kernel(const float* __restrict__ stats,
                                const float* __restrict__ gamma,
                                const float* __restrict__ beta,
                                float* __restrict__ ss) {
  const int o = threadIdx.x;
  if (o < Os) {
    const float cnt  = (float)Bs * (float)Ns * (float)Ks;
    const float mean = stats[o] / cnt;
    const float var  = stats[64 + o] / cnt - mean * mean;
    const float sc   = gamma[o] * rsqrtf(var + EPSF);
    ss[o]      = sc;
    ss[64 + o] = beta[o] - mean * sc;
  }
}

// ---------------------------------------------------------------------------
extern "C" void kernel_launch(void* const* d_in, const int* in_sizes, int n_in,
                              void* d_out, int out_size, void* d_ws,
                              size_t ws_size, hipStream_t stream) {
  const float* x     = (const float*)d_in[0];  // (B,C,N)
  const float* W     = (const float*)d_in[1];  // (O,2C)
  const float* bias  = (const float*)d_in[2];  // (O)
  const float* gamma = (const float*)d_in[3];  // (O)
  const float* beta  = (const float*)d_in[4];  // (O)

  // workspace carve-out (~11 MB)
  char*  ws    = (char*)d_ws;
  float* xT    = (float*)ws;                               // B*N*C
  float* sq    = xT + (size_t)Bs * Ns * Cs;                // B*N
  int*   idx   = (int*)(sq + (size_t)Bs * Ns);             // B*N*K
  float* stats = (float*)(idx + (size_t)Bs * Ns * Ks);     // 128
  float* ss    = stats + 128;                              // 128

  xt_sq_kernel<<<dim3(Ns / 64, Bs), 256, 0, stream>>>(x, xT, sq);
  knn_kernel<<<dim3(Ns / 32, Bs), 256, 0, stream>>>(xT, sq, idx);

  (void)hipMemsetAsync(stats, 0, 2 * Os * sizeof(float), stream);
  (void)hipMemsetAsync(d_out, 0, (size_t)out_size * sizeof(float), stream);

  const int nblk = (Bs * Ns * Ks) / 64;  // 10240
  edge_kernel<1><<<nblk, 256, 0, stream>>>(xT, idx, W, bias, stats, nullptr,
                                           nullptr);
  finalize_kernel<<<1, 64, 0, stream>>>(stats, gamma, beta, ss);
  edge_kernel<2><<<nblk, 256, 0, stream>>>(xT, idx, W, bias, nullptr, ss,
                                           (unsigned*)d_out);
}